// VariationalLinearEncoder_53120155517008
// MI455X (gfx1250) — compile-verified
//
#include <hip/hip_runtime.h>
#include <hip/hip_bf16.h>

#define IN_CH  128
#define OUT_CH 64

typedef __attribute__((ext_vector_type(2))) float v2f;
typedef __attribute__((ext_vector_type(8))) float v8f;

// ---------------------------------------------------------------------------
// Degree kernels
// ---------------------------------------------------------------------------
__global__ void k_init_deg(float* __restrict__ deg, int n) {
    int i = blockIdx.x * blockDim.x + threadIdx.x;
    if (i < n) deg[i] = 1.0f;                    // self-loop contribution
}

__global__ void k_count_deg(const long long* __restrict__ dst,
                            float* __restrict__ deg, int ne) {
    int e = blockIdx.x * blockDim.x + threadIdx.x;
    if (e < ne) atomicAdd(&deg[(int)dst[e]], 1.0f);
}

__global__ void k_rsqrt(float* __restrict__ deg, int n) {
    int i = blockIdx.x * blockDim.x + threadIdx.x;
    if (i < n) deg[i] = rsqrtf(deg[i]);          // in place: deg -> dinv
}

// ---------------------------------------------------------------------------
// Dense GEMM h = x @ W with f32 WMMA (16x16x4), fused self-loop + bias init
// of the output:  out[i][c] = h[i][c] * dinv[i]^2 + b[c]
// One block = one 16-row tile of x.  8 waves: wave>>2 selects {mu, logstd},
// wave&3 selects the 16-column tile of the 64 output channels.
//
// Fragment layouts (wave32, 32-bit data, per ISA 7.12.2):
//   A 16x4 : lane&15 = M, V0 holds K=2*(lane>>4), V1 holds K=2*(lane>>4)+1
//   B 4x16 : lane&15 = N, V0 holds K=2*(lane>>4), V1 holds K=2*(lane>>4)+1
//   C 16x16: lane&15 = N, VGPR r holds M = r + 8*(lane>>4)
// ---------------------------------------------------------------------------
__global__ __launch_bounds__(256)
void k_gemm_wmma(const float* __restrict__ x,
                 const float* __restrict__ W_mu, const float* __restrict__ b_mu,
                 const float* __restrict__ W_ls, const float* __restrict__ b_ls,
                 const float* __restrict__ dinv,
                 float* __restrict__ h_mu, float* __restrict__ h_ls,
                 float* __restrict__ out_mu, float* __restrict__ out_ls)
{
    const int wave  = threadIdx.x >> 5;   // 0..7
    const int lane  = threadIdx.x & 31;
    const int half  = lane >> 4;          // 0 or 1
    const int l16   = lane & 15;
    const int row0  = blockIdx.x * 16;
    const int which = wave >> 2;          // 0 = mu, 1 = logstd
    const int n0    = (wave & 3) * 16;

    const float* __restrict__ W = which ? W_ls : W_mu;
    const float* __restrict__ b = which ? b_ls : b_mu;
    float* __restrict__ h   = which ? h_ls  : h_mu;
    float* __restrict__ out = which ? out_ls : out_mu;

    const float* __restrict__ arow = x + (size_t)(row0 + l16) * IN_CH;

    v8f c = {};
    #pragma unroll 4
    for (int k0 = 0; k0 < IN_CH; k0 += 4) {
        v2f a;
        const float* ap = arow + k0 + half * 2;
        a.x = ap[0];
        a.y = ap[1];

        v2f bb;
        const float* bp = W + (size_t)(k0 + half * 2) * OUT_CH + n0 + l16;
        bb.x = bp[0];
        bb.y = bp[OUT_CH];

        c = __builtin_amdgcn_wmma_f32_16x16x4_f32(
                /*neg_a=*/false, a, /*neg_b=*/false, bb,
                /*c_mod=*/(short)0, c, /*reuse_a=*/false, /*reuse_b=*/false);
    }

    const int col = n0 + l16;
    const float bias = b[col];
    #pragma unroll
    for (int r = 0; r < 8; ++r) {
        const int row = row0 + r + half * 8;
        const float hv = c[r];
        const float di = dinv[row];
        h[(size_t)row * OUT_CH + col]   = hv;
        out[(size_t)row * OUT_CH + col] = hv * di * di + bias;
    }
}

// ---------------------------------------------------------------------------
// Edge scatter:  out[dst] += h[src] * dinv[src] * dinv[dst]
// 16 threads per edge; each thread owns a 4-channel float4 chunk and handles
// both convs.  Gathers are coalesced float4; scatters are f32 atomics into
// L2-resident output (both 25.6 MB outputs fit the 192 MB L2).
// ---------------------------------------------------------------------------
__global__ __launch_bounds__(256)
void k_edge_scatter(const long long* __restrict__ ei,   // [2, ne]
                    const float* __restrict__ dinv,
                    const float* __restrict__ h_mu,
                    const float* __restrict__ h_ls,
                    float* __restrict__ out_mu,
                    float* __restrict__ out_ls,
                    int ne)
{
    const int gid = blockIdx.x * blockDim.x + threadIdx.x;
    const int e   = gid >> 4;
    if (e >= ne) return;
    const int c4 = (gid & 15) * 4;

    const int s = (int)ei[e];
    const int d = (int)ei[(size_t)ne + e];
    const float norm = dinv[s] * dinv[d];

    const float4 hm = *(const float4*)(h_mu + (size_t)s * OUT_CH + c4);
    const float4 hl = *(const float4*)(h_ls + (size_t)s * OUT_CH + c4);

    float* om = out_mu + (size_t)d * OUT_CH + c4;
    float* ol = out_ls + (size_t)d * OUT_CH + c4;

    atomicAdd(om + 0, hm.x * norm);
    atomicAdd(om + 1, hm.y * norm);
    atomicAdd(om + 2, hm.z * norm);
    atomicAdd(om + 3, hm.w * norm);
    atomicAdd(ol + 0, hl.x * norm);
    atomicAdd(ol + 1, hl.y * norm);
    atomicAdd(ol + 2, hl.z * norm);
    atomicAdd(ol + 3, hl.w * norm);
}

// ---------------------------------------------------------------------------
// Launch
// ---------------------------------------------------------------------------
extern "C" void kernel_launch(void* const* d_in, const int* in_sizes, int n_in,
                              void* d_out, int out_size, void* d_ws, size_t ws_size,
                              hipStream_t stream)
{
    const float*     x    = (const float*)d_in[0];
    const float*     W_mu = (const float*)d_in[1];
    const float*     b_mu = (const float*)d_in[2];
    const float*     W_ls = (const float*)d_in[3];
    const float*     b_ls = (const float*)d_in[4];
    const long long* ei   = (const long long*)d_in[5];   // int64 [2, ne]

    const int n_nodes = in_sizes[0] / IN_CH;   // 100000
    const int n_edges = in_sizes[5] / 2;       // 1600000

    // workspace layout: h_mu | h_ls | dinv
    float* h_mu = (float*)d_ws;
    float* h_ls = h_mu + (size_t)n_nodes * OUT_CH;
    float* dinv = h_ls + (size_t)n_nodes * OUT_CH;

    // output layout: mu | logstd
    float* out_mu = (float*)d_out;
    float* out_ls = out_mu + (size_t)n_nodes * OUT_CH;

    const int TB = 256;

    // 1) degree with self-loops -> dinv
    k_init_deg<<<(n_nodes + TB - 1) / TB, TB, 0, stream>>>(dinv, n_nodes);
    k_count_deg<<<(n_edges + TB - 1) / TB, TB, 0, stream>>>(ei + n_edges, dinv, n_edges);
    k_rsqrt<<<(n_nodes + TB - 1) / TB, TB, 0, stream>>>(dinv, n_nodes);

    // 2) WMMA GEMM for both weight matrices + self-loop/bias output init
    const int row_tiles = (n_nodes + 15) / 16;   // 6250 (exact)
    k_gemm_wmma<<<row_tiles, 256, 0, stream>>>(x, W_mu, b_mu, W_ls, b_ls, dinv,
                                               h_mu, h_ls, out_mu, out_ls);

    // 3) edge aggregation (16 threads/edge)
    const long long work = (long long)n_edges * 16;
    k_edge_scatter<<<(int)((work + TB - 1) / TB), TB, 0, stream>>>(
        ei, dinv, h_mu, h_ls, out_mu, out_ls, n_edges);
}